// TransformerBlock_33698313404795
// MI455X (gfx1250) — compile-verified
//
#include <hip/hip_runtime.h>

// ---------------- problem constants (match reference) ----------------
constexpr int kB     = 32;
constexpr int kDIM   = 4096;
constexpr int kNH    = 32;
constexpr int kNKV   = 8;
constexpr int kHD    = 128;
constexpr int kINTER = 14336;
constexpr int kPS    = 16;
constexpr int kQKV   = (kNH + 2 * kNKV) * kHD;   // 6144
constexpr int kG     = kNH / kNKV;               // 4

constexpr int kKC  = 64;   // K chunk staged per double-buffer slot
constexpr int kLDX = 68;   // LDS row stride (floats): conflict-free, 16B aligned
constexpr int kNT  = 128;  // N tile per block (2x 16-wide tiles per wave)

typedef __attribute__((ext_vector_type(2))) float v2f;
typedef __attribute__((ext_vector_type(8))) float v8f;
typedef __attribute__((ext_vector_type(4))) int   v4i;

// ---------------- WMMA fp32 16x16x4 wrapper ----------------
__device__ inline v8f wmma_f32(v2f a, v2f b, v8f c) {
#if defined(__HIP_DEVICE_COMPILE__) && __has_builtin(__builtin_amdgcn_wmma_f32_16x16x4_f32)
    return __builtin_amdgcn_wmma_f32_16x16x4_f32(false, a, false, b, (short)0, c,
                                                 false, false);
#else
    return c;
#endif
}

// ---------------- CDNA5 async global->LDS copy (16B per lane) ----------------
__device__ inline void async_copy16(const float* gsrc, float* ldst) {
#if defined(__HIP_DEVICE_COMPILE__) && __has_builtin(__builtin_amdgcn_global_load_async_to_lds_b128)
    __builtin_amdgcn_global_load_async_to_lds_b128(
        (__attribute__((address_space(1))) v4i*)gsrc,
        (__attribute__((address_space(3))) v4i*)ldst, 0, 0);
#else
    *(float4*)ldst = *(const float4*)gsrc;   // synchronous fallback
#endif
}

template <int N>
__device__ inline void wait_async_le() {
#if defined(__HIP_DEVICE_COMPILE__)
#if __has_builtin(__builtin_amdgcn_s_wait_asynccnt)
    __builtin_amdgcn_s_wait_asynccnt(N);
#elif __has_builtin(__builtin_amdgcn_global_load_async_to_lds_b128)
    asm volatile("s_wait_asynccnt %0" ::"i"(N) : "memory");
#endif
#endif
}

// ---------------- RMSNorm: one block per row ----------------
__global__ __launch_bounds__(256) void rmsnorm_kernel(const float* __restrict__ x,
                                                      const float* __restrict__ w,
                                                      float* __restrict__ out, int dim) {
    const int row = blockIdx.x;
    const float* xr = x + (size_t)row * dim;
    float ss = 0.f;
    for (int i = threadIdx.x; i < dim; i += 256) {
        float v = xr[i];
        ss += v * v;
    }
#pragma unroll
    for (int off = 16; off >= 1; off >>= 1) ss += __shfl_xor(ss, off, 32);
    __shared__ float red[8];
    const int wave = threadIdx.x >> 5, lane = threadIdx.x & 31;
    if (lane == 0) red[wave] = ss;
    __syncthreads();
    float tot = 0.f;
#pragma unroll
    for (int i = 0; i < 8; i++) tot += red[i];
    const float rs = rsqrtf(tot / (float)dim + 1e-5f);
    for (int i = threadIdx.x; i < dim; i += 256)
        out[(size_t)row * dim + i] = xr[i] * rs * w[i];
}

// ---------------- split-K GEMM: part[seg] = X[32,Kseg] @ W[N,Kseg]^T ----------------
// Grid: (N/128, nseg). Block: 256 threads = 8 waves; wave owns 16(M) x 32(N) via 2 C tiles.
// Double-buffered LDS filled with async global->LDS; ASYNCcnt-throttled pipeline.
__global__ __launch_bounds__(256) void gemm32_kernel(const float* __restrict__ X,
                                                     const float* __restrict__ W,
                                                     float* __restrict__ part,
                                                     int K, int N) {
    __shared__ float Xs[2][32][kLDX];
    __shared__ float Ws[2][kNT][kLDX];
    const int tid  = threadIdx.x;
    const int wave = tid >> 5;
    const int lane = tid & 31;
    const int nblk = blockIdx.x * kNT;
    const int seg  = blockIdx.y;
    const int segK = K / gridDim.y;          // multiple of kKC by construction
    const int kbase = seg * segK;
    const int wm   = (wave & 1) * 16;        // wave M tile: 0 / 16
    const int wn   = (wave >> 1) * 32;       // wave N tiles: wn, wn+16
    const int srow = tid >> 4;               // 0..15 staging row
    const int scol = (tid & 15) * 4;         // 0..60 staging col (float4)

    const int nch = segK / kKC;

    // stage chunk `ci` (local index) into buffer buf
    auto stage = [&](int buf, int ci) {
        const int k0 = kbase + ci * kKC;
#pragma unroll
        for (int j = 0; j < 2; j++)          // X: 32 x 64
            async_copy16(X + (size_t)(srow + 16 * j) * K + k0 + scol,
                         &Xs[buf][srow + 16 * j][scol]);
#pragma unroll
        for (int j = 0; j < 8; j++)          // W: 128 x 64
            async_copy16(W + (size_t)(nblk + srow + 16 * j) * K + k0 + scol,
                         &Ws[buf][srow + 16 * j][scol]);
    };

    v8f c0 = {0.f, 0.f, 0.f, 0.f, 0.f, 0.f, 0.f, 0.f};
    v8f c1 = {0.f, 0.f, 0.f, 0.f, 0.f, 0.f, 0.f, 0.f};
    const int l15 = lane & 15;
    const int hi  = lane >> 4;   // lanes 16-31 hold K+2 per f32 WMMA A/B layout

    stage(0, 0);
    for (int i = 0; i < nch; i++) {
        const int cur = i & 1;
        if (i + 1 < nch) {       // prefetch next chunk, then wait for current (10 in flight)
            stage(cur ^ 1, i + 1);
            wait_async_le<10>();
        } else {
            wait_async_le<0>();
        }
        __syncthreads();
#pragma unroll
        for (int s = 0; s < 16; s++) {
            const int kk = s * 4 + hi * 2;
            v2f a, b0, b1;
            a.x  = Xs[cur][wm + l15][kk];
            a.y  = Xs[cur][wm + l15][kk + 1];
            b0.x = Ws[cur][wn + l15][kk];
            b0.y = Ws[cur][wn + l15][kk + 1];
            b1.x = Ws[cur][wn + 16 + l15][kk];
            b1.y = Ws[cur][wn + 16 + l15][kk + 1];
            c0 = wmma_f32(a, b0, c0);
            c1 = wmma_f32(a, b1, c1);
        }
        __syncthreads();   // all waves done reading buf[cur] before it is refilled
    }

    // partial store: lane holds N = l15 (+0 / +16 tile), VGPR r holds M = r (+8 hi)
    const int n0 = nblk + wn + l15;
#pragma unroll
    for (int r = 0; r < 8; r++) {
        const int m = wm + r + hi * 8;
        float* dst = part + ((size_t)seg * 32 + m) * N + n0;
        dst[0]  = c0[r];
        dst[16] = c1[r];
    }
}

// ---------------- split-K reduce + epilogue ----------------
// mode 0: plain  mode 1: += res[m*N+n]  mode 2: silu(sum) * stem_buf[stem_ids[m]*N+n]
__global__ __launch_bounds__(256) void reduce_kernel(const float* __restrict__ part,
                                                     int nseg,
                                                     const float* __restrict__ res,
                                                     const float* __restrict__ stem_buf,
                                                     const int* __restrict__ stem_ids,
                                                     float* __restrict__ out,
                                                     int N, int mode) {
    const size_t idx4  = ((size_t)blockIdx.x * 256 + threadIdx.x) * 4;
    const size_t total = (size_t)32 * N;
    if (idx4 >= total) return;
    float4 s = make_float4(0.f, 0.f, 0.f, 0.f);
    for (int sg = 0; sg < nseg; sg++) {
        const float4 p = *(const float4*)(part + (size_t)sg * total + idx4);
        s.x += p.x; s.y += p.y; s.z += p.z; s.w += p.w;
    }
    const int m = (int)(idx4 / N);
    const int n = (int)(idx4 % N);
    if (mode == 1) {
        const float4 r = *(const float4*)(res + idx4);
        s.x += r.x; s.y += r.y; s.z += r.z; s.w += r.w;
    } else if (mode == 2) {
        const float4 st = *(const float4*)(stem_buf + (size_t)stem_ids[m] * N + n);
        s.x = (s.x / (1.f + __expf(-s.x))) * st.x;
        s.y = (s.y / (1.f + __expf(-s.y))) * st.y;
        s.z = (s.z / (1.f + __expf(-s.z))) * st.z;
        s.w = (s.w / (1.f + __expf(-s.w))) * st.w;
    }
    *(float4*)(out + idx4) = s;
}

// ---------------- RoPE: one block = one (batch, head); 64 threads ----------------
__global__ __launch_bounds__(64) void rope_kernel(const float* __restrict__ qkv,
                                                  const int* __restrict__ offsets,
                                                  float* __restrict__ qout,
                                                  float* __restrict__ kout,
                                                  float* __restrict__ vout) {
    const int b = blockIdx.x / (kNH + kNKV);
    const int r = blockIdx.x % (kNH + kNKV);
    const int i = threadIdx.x;  // 0..63 (half head dim)
    const float pos = (float)offsets[b];
    const float inv = __powf(10000.f, -(float)i / 64.f);
    float sn, cs;
    __sincosf(pos * inv, &sn, &cs);
    if (r < kNH) {
        const float* src = qkv + (size_t)b * kQKV + r * kHD;
        const float x1 = src[i], x2 = src[i + 64];
        float* dst = qout + ((size_t)b * kNH + r) * kHD;
        dst[i]      = x1 * cs - x2 * sn;
        dst[i + 64] = x1 * sn + x2 * cs;
    } else {
        const int h = r - kNH;
        const float* ks = qkv + (size_t)b * kQKV + (kNH + h) * kHD;
        const float* vs = qkv + (size_t)b * kQKV + (kNH + kNKV + h) * kHD;
        const float x1 = ks[i], x2 = ks[i + 64];
        float* kd = kout + ((size_t)b * kNKV + h) * kHD;
        float* vd = vout + ((size_t)b * kNKV + h) * kHD;
        kd[i]      = x1 * cs - x2 * sn;
        kd[i + 64] = x1 * sn + x2 * cs;
        vd[i]      = vs[i];
        vd[i + 64] = vs[i + 64];
    }
}

// ---------------- Flash attention: one block = one (batch, kv-head) ----------------
__global__ __launch_bounds__(256) void attn_kernel(const float* __restrict__ q,
                                                   const float* __restrict__ knew,
                                                   const float* __restrict__ vnew,
                                                   const float* __restrict__ pk,
                                                   const float* __restrict__ pv,
                                                   const int* __restrict__ page_indices,
                                                   const int* __restrict__ page_indptr,
                                                   const int* __restrict__ lastlen,
                                                   float* __restrict__ o) {
    const int b    = blockIdx.x / kNKV;
    const int h    = blockIdx.x % kNKV;
    const int tid  = threadIdx.x;
    const int wave = tid >> 5;
    const int lane = tid & 31;
    const int p0   = page_indptr[b];
    const int npg  = page_indptr[b + 1] - p0;
    const int len  = (npg - 1) * kPS + lastlen[b];

    __shared__ float qs[kG][kHD];
    __shared__ float mL[8][kG], lL[8][kG];
    __shared__ float accL[8][kG][kHD];

    const float scale = 0.08838834764831845f;  // 1/sqrt(128)
    for (int i = tid; i < kG * kHD; i += 256) {
        const int g = i >> 7, d = i & 127;
        qs[g][d] = q[((size_t)b * kNH + h * kG + g) * kHD + d] * scale;
    }
    __syncthreads();

    float4 qv[kG];
#pragma unroll
    for (int g = 0; g < kG; g++) qv[g] = *(const float4*)&qs[g][4 * lane];

    float m[kG], l[kG];
    float4 acc[kG];
#pragma unroll
    for (int g = 0; g < kG; g++) {
        m[g] = -1e30f;
        l[g] = 0.f;
        acc[g] = make_float4(0.f, 0.f, 0.f, 0.f);
    }

    for (int p = wave; p < len; p += 8) {
        const float* kvec;
        const float* vvec;
        if (p == len - 1) {  // freshly appended token (inputs immutable)
            kvec = knew + ((size_t)b * kNKV + h) * kHD;
            vvec = vnew + ((size_t)b * kNKV + h) * kHD;
        } else {
            const int pg = page_indices[p0 + (p >> 4)];
            const int sl = p & 15;
            const size_t base = (((size_t)pg * kPS + sl) * kNKV + h) * kHD;
            kvec = pk + base;
            vvec = pv + base;
        }
        const float4 kv4 = *(const float4*)(kvec + 4 * lane);
        float s[kG];
#pragma unroll
        for (int g = 0; g < kG; g++) {
            float d = qv[g].x * kv4.x + qv[g].y * kv4.y + qv[g].z * kv4.z + qv[g].w * kv4.w;
#pragma unroll
            for (int off = 16; off >= 1; off >>= 1) d += __shfl_xor(d, off, 32);
            s[g] = d;
        }
        const float4 vv4 = *(const float4*)(vvec + 4 * lane);
#pragma unroll
        for (int g = 0; g < kG; g++) {
            const float mn = fmaxf(m[g], s[g]);
            const float sc = __expf(m[g] - mn);
            const float pe = __expf(s[g] - mn);
            m[g] = mn;
            l[g] = l[g] * sc + pe;
            acc[g].x = acc[g].x * sc + pe * vv4.x;
            acc[g].y = acc[g].y * sc + pe * vv4.y;
            acc[g].z = acc[g].z * sc + pe * vv4.z;
            acc[g].w = acc[g].w * sc + pe * vv4.w;
        }
    }

    if (lane == 0) {
#pragma unroll
        for (int g = 0; g < kG; g++) {
            mL[wave][g] = m[g];
            lL[wave][g] = l[g];
        }
    }
#pragma unroll
    for (int g = 0; g < kG; g++) *(float4*)&accL[wave][g][4 * lane] = acc[g];
    __syncthreads();

    if (wave < kG) {
        const int g = wave;
        float M = -1e30f;
#pragma unroll
        for (int w = 0; w < 8; w++) M = fmaxf(M, mL[w][g]);
        float L = 0.f;
#pragma unroll
        for (int w = 0; w < 8; w++) L += lL[w][g] * __expf(mL[w][g] - M);
        const float invL = 1.f / L;
        float ox = 0.f, oy = 0.f, oz = 0.f, ow = 0.f;
#pragma unroll
        for (int w = 0; w < 8; w++) {
            const float f = __expf(mL[w][g] - M) * invL;
            const float4 a = *(const float4*)&accL[w][g][4 * lane];
            ox += a.x * f; oy += a.y * f; oz += a.z * f; ow += a.w * f;
        }
        float* dst = o + (size_t)b * kNH * kHD + (h * kG + g) * kHD + 4 * lane;
        dst[0] = ox; dst[1] = oy; dst[2] = oz; dst[3] = ow;
    }
}

// ---------------- host-side orchestration ----------------
extern "C" void kernel_launch(void* const* d_in, const int* in_sizes, int n_in,
                              void* d_out, int out_size, void* d_ws, size_t ws_size,
                              hipStream_t stream) {
    (void)in_sizes; (void)n_in; (void)out_size; (void)ws_size;
    const float* x           = (const float*)d_in[0];
    const float* buffer      = (const float*)d_in[1];
    const float* paged_k     = (const float*)d_in[2];
    const float* paged_v     = (const float*)d_in[3];
    const float* wqkv        = (const float*)d_in[4];
    const float* wo          = (const float*)d_in[5];
    const float* w1          = (const float*)d_in[6];
    const float* w2          = (const float*)d_in[7];
    const float* attn_norm_w = (const float*)d_in[8];
    const float* ffn_norm_w  = (const float*)d_in[9];
    const int*   buffer_ids  = (const int*)d_in[10];
    const int*   offsets     = (const int*)d_in[11];
    const int*   kv_page_indices = (const int*)d_in[13];
    const int*   kv_page_indptr  = (const int*)d_in[14];
    const int*   kv_page_lastlen = (const int*)d_in[15];
    float* out = (float*)d_out;

    float* ws = (float*)d_ws;
    float* xa     = ws; ws += (size_t)kB * kDIM;
    float* qkv    = ws; ws += (size_t)kB * kQKV;
    float* qro    = ws; ws += (size_t)kB * kNH * kHD;
    float* knew   = ws; ws += (size_t)kB * kNKV * kHD;
    float* vnew   = ws; ws += (size_t)kB * kNKV * kHD;
    float* ob     = ws; ws += (size_t)kB * kNH * kHD;
    float* h      = ws; ws += (size_t)kB * kDIM;
    float* hf     = ws; ws += (size_t)kB * kDIM;
    float* hidden = ws; ws += (size_t)kB * kINTER;
    float* part   = ws; ws += (size_t)8 * kB * kDIM;   // max nseg*32*N = 8*32*4096

    // 1) attn rmsnorm
    rmsnorm_kernel<<<kB, 256, 0, stream>>>(x, attn_norm_w, xa, kDIM);
    // 2) QKV projection: 48 N-tiles x 4 K-segments
    gemm32_kernel<<<dim3(kQKV / kNT, 4), 256, 0, stream>>>(xa, wqkv, part, kDIM, kQKV);
    reduce_kernel<<<(kB * kQKV) / 1024, 256, 0, stream>>>(part, 4, nullptr, nullptr,
                                                          nullptr, qkv, kQKV, 0);
    // 3) RoPE on q and k_new, copy v_new
    rope_kernel<<<kB * (kNH + kNKV), 64, 0, stream>>>(qkv, offsets, qro, knew, vnew);
    // 4) paged GQA flash attention
    attn_kernel<<<kB * kNKV, 256, 0, stream>>>(qro, knew, vnew, paged_k, paged_v,
                                               kv_page_indices, kv_page_indptr,
                                               kv_page_lastlen, ob);
    // 5) output projection + residual: 32 N-tiles x 8 K-segments
    gemm32_kernel<<<dim3(kDIM / kNT, 8), 256, 0, stream>>>(ob, wo, part, kNH * kHD, kDIM);
    reduce_kernel<<<(kB * kDIM) / 1024, 256, 0, stream>>>(part, 8, x, nullptr, nullptr,
                                                          h, kDIM, 1);
    // 6) ffn rmsnorm
    rmsnorm_kernel<<<kB, 256, 0, stream>>>(h, ffn_norm_w, hf, kDIM);
    // 7) gate proj (112 N-tiles x 2 K-segments) + silu * stem epilogue
    gemm32_kernel<<<dim3(kINTER / kNT, 2), 256, 0, stream>>>(hf, w1, part, kDIM, kINTER);
    reduce_kernel<<<(kB * kINTER) / 1024, 256, 0, stream>>>(part, 2, nullptr, buffer,
                                                            buffer_ids, hidden, kINTER, 2);
    // 8) down proj (32 N-tiles x 8 K-segments) + residual -> out
    gemm32_kernel<<<dim3(kDIM / kNT, 8), 256, 0, stream>>>(hidden, w2, part, kINTER, kDIM);
    reduce_kernel<<<(kB * kDIM) / 1024, 256, 0, stream>>>(part, 8, h, nullptr, nullptr,
                                                          out, kDIM, 1);
}